// TopKAutoencodeInhibitor_63642825392618
// MI455X (gfx1250) — compile-verified
//
#include <hip/hip_runtime.h>
#include <stdint.h>

// Problem constants
#define NROWS 8192
#define DDIM  512
#define MEXP  2048
#define BDIM  4
#define KTOP  8
#define MB    (MEXP*BDIM)      // 8192 floats per h row
// d_out layout (floats): h_sparse [0,262144), idx [262144,327680), scalars [327680,327687)
#define OUT_IDX_OFF   262144
#define OUT_SCAL_OFF  327680

typedef float v2f __attribute__((ext_vector_type(2)));
typedef float v8f __attribute__((ext_vector_type(8)));

// ---------------- init: zero the integer histogram ----------------
__global__ void zero_counts_kernel(unsigned* __restrict__ counts) {
    int t = threadIdx.x;
    #pragma unroll
    for (int i = 0; i < MEXP / 256; ++i) counts[t + (i << 8)] = 0u;
}

// ---------------- V (D,M,B) -> Vt (M*B, D), tiled LDS transpose ----------------
__global__ __launch_bounds__(256) void transpose_v_kernel(const float* __restrict__ V,
                                                          float* __restrict__ Vt) {
    __shared__ float tile[32][33];
    int bx = blockIdx.x * 32;           // column block (mb index)
    int by = blockIdx.y * 32;           // row block (d index)
    int tx = threadIdx.x, ty = threadIdx.y;   // 32 x 8
    #pragma unroll
    for (int j = 0; j < 32; j += 8)
        tile[ty + j][tx] = V[(size_t)(by + ty + j) * MB + bx + tx];
    __syncthreads();
    #pragma unroll
    for (int j = 0; j < 32; j += 8)
        Vt[(size_t)(bx + ty + j) * DDIM + by + tx] = tile[tx][ty + j];
}

// ---------------- energy + top-8 + column partial sums ----------------
__global__ __launch_bounds__(256) void energy_topk_kernel(
    const float* __restrict__ h, float* __restrict__ dout,
    int* __restrict__ idx_int, float* __restrict__ partial,
    float* __restrict__ captured_row, unsigned* __restrict__ counts) {
    __shared__ unsigned long long warpbest[8];
    __shared__ unsigned long long winner;
    __shared__ int   mk_sh[KTOP];
    __shared__ float vk_sh[KTOP];

    int tid = threadIdx.x, lane = tid & 31, wid = tid >> 5;
    int wg = blockIdx.x;
    float colacc[8];
    #pragma unroll
    for (int i = 0; i < 8; ++i) colacc[i] = 0.0f;

    for (int r = 0; r < 32; ++r) {
        int n = wg * 32 + r;
        const float4* hrow = (const float4*)(h + (size_t)n * MB);
        float e[8];
        #pragma unroll
        for (int i = 0; i < 8; ++i) {
            float4 v = hrow[tid + (i << 8)];                // coalesced 16B/lane
            float en = v.x * v.x + v.y * v.y + v.z * v.z + v.w * v.w;
            e[i] = en;
            colacc[i] += en;
        }
        if (r < 31)   // warm next row into caches -> global_prefetch_b8
            __builtin_prefetch(h + (size_t)(n + 1) * MB + tid * 32, 0, 0);

        // 8 rounds of argmax; key = (fbits<<32)|~m  (energy >= 0, lower idx wins ties)
        for (int k = 0; k < KTOP; ++k) {
            unsigned long long key = 0ull;
            #pragma unroll
            for (int i = 0; i < 8; ++i) {
                float v = e[i];
                unsigned m = (unsigned)(tid + (i << 8));
                unsigned long long c = (v < 0.0f) ? 0ull
                    : (((unsigned long long)__float_as_uint(v) << 32) |
                       (unsigned long long)(0xFFFFFFFFu - m));
                key = (c > key) ? c : key;
            }
            #pragma unroll
            for (int s = 16; s >= 1; s >>= 1) {
                unsigned long long o = __shfl_xor(key, s, 32);
                key = (o > key) ? o : key;
            }
            if (lane == 0) warpbest[wid] = key;
            __syncthreads();
            if (tid == 0) {
                unsigned long long bk = warpbest[0];
                for (int w = 1; w < 8; ++w) if (warpbest[w] > bk) bk = warpbest[w];
                winner = bk;
                unsigned m = 0xFFFFFFFFu - (unsigned)(bk & 0xFFFFFFFFull);
                mk_sh[k] = (int)m;
                vk_sh[k] = __uint_as_float((unsigned)(bk >> 32));
            }
            __syncthreads();
            unsigned long long bk = winner;
            unsigned m = 0xFFFFFFFFu - (unsigned)(bk & 0xFFFFFFFFull);
            // owner zaps its copy (unrolled to keep e[] in registers)
            if ((m & 255u) == (unsigned)tid) {
                unsigned hi = m >> 8;
                #pragma unroll
                for (int i = 0; i < 8; ++i) if (hi == (unsigned)i) e[i] = -1.0f;
            }
        }
        // outputs for this row
        if (tid < 32) {  // h_sparse gather (K*B = 32 values)
            int k = tid >> 2;
            dout[(size_t)n * 32 + tid] = h[(size_t)n * MB + (size_t)mk_sh[k] * 4 + (tid & 3)];
        }
        if (tid < KTOP) {
            dout[OUT_IDX_OFF + (size_t)n * KTOP + tid] = (float)mk_sh[tid];
            idx_int[n * KTOP + tid] = mk_sh[tid];
        }
        if (tid == 0) {
            float s = 0.0f;
            #pragma unroll
            for (int k = 0; k < KTOP; ++k) s += vk_sh[k];
            captured_row[n] = s;
            for (int k = 0; k < KTOP; ++k) atomicAdd(&counts[mk_sh[k]], 1u);
        }
        __syncthreads();
    }
    #pragma unroll
    for (int i = 0; i < 8; ++i)
        partial[(size_t)wg * MEXP + tid + (i << 8)] = colacc[i];
}

// ---------------- x_hat via V_WMMA_F32_16X16X4_F32 + residual energies ----------------
__global__ __launch_bounds__(256) void xhat_wmma_kernel(
    const float* __restrict__ x_flat, const float* __restrict__ hsp,
    const int* __restrict__ idx_int, const float* __restrict__ Vt,
    float* __restrict__ uncap_row, float* __restrict__ recon_row) {
    __shared__ float c_sh[32];
    __shared__ int   rows_sh[32];
    __shared__ float red[16];

    int tid = threadIdx.x, lane = tid & 31, wid = tid >> 5;
    int n = blockIdx.x;
    if (tid < 32) {
        c_sh[tid] = hsp[(size_t)n * 32 + tid];
        rows_sh[tid] = idx_int[n * KTOP + (tid >> 2)] * BDIM + (tid & 3);
    }
    __syncthreads();

    v8f acc0 = {}, acc1 = {}, acc2 = {}, acc3 = {};
    int lh  = (lane >> 4) & 1;     // which K-pair this half-wave holds
    int col = lane & 15;
    int dbase = wid * 64;

    #pragma unroll
    for (int chunk = 0; chunk < 8; ++chunk) {
        int kb = chunk * 4 + lh * 2;
        v2f a; a.x = c_sh[kb]; a.y = c_sh[kb + 1];     // broadcast coeffs to all M rows
        int r0 = rows_sh[kb] * DDIM, r1 = rows_sh[kb + 1] * DDIM;
        v2f b0; b0.x = Vt[r0 + dbase + col];        b0.y = Vt[r1 + dbase + col];
        v2f b1; b1.x = Vt[r0 + dbase + 16 + col];   b1.y = Vt[r1 + dbase + 16 + col];
        v2f b2; b2.x = Vt[r0 + dbase + 32 + col];   b2.y = Vt[r1 + dbase + 32 + col];
        v2f b3; b3.x = Vt[r0 + dbase + 48 + col];   b3.y = Vt[r1 + dbase + 48 + col];
        acc0 = __builtin_amdgcn_wmma_f32_16x16x4_f32(false, a, false, b0, (short)0, acc0, false, false);
        acc1 = __builtin_amdgcn_wmma_f32_16x16x4_f32(false, a, false, b1, (short)0, acc1, false, false);
        acc2 = __builtin_amdgcn_wmma_f32_16x16x4_f32(false, a, false, b2, (short)0, acc2, false, false);
        acc3 = __builtin_amdgcn_wmma_f32_16x16x4_f32(false, a, false, b3, (short)0, acc3, false, false);
    }

    // all 16 M-rows identical -> element 0 of each lane's accumulator is x_hat[d]
    float lu = 0.0f, lr = 0.0f;
    if (lane < 16) {
        float xh, x, rr;
        xh = acc0[0]; x = x_flat[(size_t)n * DDIM + dbase + col];      rr = x - xh; lu += rr * rr; lr += xh * xh;
        xh = acc1[0]; x = x_flat[(size_t)n * DDIM + dbase + 16 + col]; rr = x - xh; lu += rr * rr; lr += xh * xh;
        xh = acc2[0]; x = x_flat[(size_t)n * DDIM + dbase + 32 + col]; rr = x - xh; lu += rr * rr; lr += xh * xh;
        xh = acc3[0]; x = x_flat[(size_t)n * DDIM + dbase + 48 + col]; rr = x - xh; lu += rr * rr; lr += xh * xh;
    }
    #pragma unroll
    for (int s = 16; s >= 1; s >>= 1) { lu += __shfl_xor(lu, s, 32); lr += __shfl_xor(lr, s, 32); }
    if (lane == 0) { red[wid] = lu; red[8 + wid] = lr; }
    __syncthreads();
    if (tid == 0) {
        float su = 0.0f, sr = 0.0f;
        #pragma unroll
        for (int w = 0; w < 8; ++w) { su += red[w]; sr += red[8 + w]; }
        uncap_row[n] = su;
        recon_row[n] = sr;
    }
}

// ---------------- deterministic column-sum reduction -> avg energy per expert ----------------
__global__ __launch_bounds__(256) void colsum_reduce_kernel(const float* __restrict__ partial,
                                                            float* __restrict__ avg) {
    int m = blockIdx.x * 256 + threadIdx.x;
    float s = 0.0f;
    for (int c = 0; c < 256; ++c) s += partial[(size_t)c * MEXP + m];
    avg[m] = s / (float)NROWS;
}

// ---------------- final scalars ----------------
__device__ float block_sum(float v, float* sred) {
    int tid = threadIdx.x;
    sred[tid] = v; __syncthreads();
    for (int s = 128; s > 0; s >>= 1) {
        if (tid < s) sred[tid] += sred[tid + s];
        __syncthreads();
    }
    float r = sred[0]; __syncthreads();
    return r;
}

__global__ __launch_bounds__(256) void finalize_kernel(
    const float* __restrict__ captured_row, const float* __restrict__ uncap_row,
    const float* __restrict__ recon_row, const float* __restrict__ avg,
    const unsigned* __restrict__ counts, float* __restrict__ out) {
    __shared__ float sred[256];
    int tid = threadIdx.x;
    float s;

    s = 0.0f; for (int i = 0; i < 32; ++i) s += captured_row[tid + (i << 8)];
    float cap = block_sum(s, sred);
    s = 0.0f; for (int i = 0; i < 32; ++i) s += uncap_row[tid + (i << 8)];
    float unc = block_sum(s, sred);
    s = 0.0f; for (int i = 0; i < 32; ++i) s += recon_row[tid + (i << 8)];
    float rec = block_sum(s, sred);

    s = 0.0f; for (int i = 0; i < 8; ++i) s += avg[tid + (i << 8)];
    float asum = block_sum(s, sred);
    float denom = fmaxf(asum, 1e-8f);
    s = 0.0f;
    for (int i = 0; i < 8; ++i) {
        float p = fmaxf(avg[tid + (i << 8)] / denom, 1e-8f);
        s += p * logf(p);
    }
    float plogp = block_sum(s, sred);
    float entropy = -plogp / logf((float)MEXP);

    // expected = K/M*N = 32
    float lo = 0.0f, nd = 0.0f;
    for (int i = 0; i < 8; ++i) {
        float c = (float)counts[tid + (i << 8)];
        lo += (c <= 3.2f)  ? 1.0f : 0.0f;
        nd += (c <= 0.32f) ? 1.0f : 0.0f;
    }
    lo = block_sum(lo, sred);
    nd = block_sum(nd, sred);

    if (tid == 0) {
        float capm = cap / (float)NROWS;
        float recm = rec / (float)NROWS;
        float uncm = unc / (float)NROWS;
        out[OUT_SCAL_OFF + 0] = capm;
        out[OUT_SCAL_OFF + 1] = recm;
        out[OUT_SCAL_OFF + 2] = uncm;
        out[OUT_SCAL_OFF + 3] = entropy;
        out[OUT_SCAL_OFF + 4] = uncm + 0.01f * (1.0f - entropy);
        out[OUT_SCAL_OFF + 5] = lo;
        out[OUT_SCAL_OFF + 6] = nd;
    }
}

extern "C" void kernel_launch(void* const* d_in, const int* in_sizes, int n_in,
                              void* d_out, int out_size, void* d_ws, size_t ws_size,
                              hipStream_t stream) {
    const float* x_flat = (const float*)d_in[0];   // (N, D)
    const float* h_all  = (const float*)d_in[1];   // (N, M, B)
    const float* V      = (const float*)d_in[2];   // (D, M, B)
    float* out = (float*)d_out;

    // workspace layout (~19.3 MB total)
    char* ws = (char*)d_ws;
    float*    Vt           = (float*)(ws);                 // 16,777,216 B
    float*    partial      = (float*)(ws + 16777216);      //  2,097,152 B (256 x 2048)
    float*    captured_row = (float*)(ws + 18874368);      //     32,768 B
    float*    uncap_row    = (float*)(ws + 18907136);      //     32,768 B
    float*    recon_row    = (float*)(ws + 18939904);      //     32,768 B
    int*      idx_int      = (int*)  (ws + 18972672);      //    262,144 B
    unsigned* counts       = (unsigned*)(ws + 19234816);   //      8,192 B
    float*    avg          = (float*)(ws + 19243008);      //      8,192 B

    zero_counts_kernel<<<1, 256, 0, stream>>>(counts);
    transpose_v_kernel<<<dim3(MB / 32, DDIM / 32), dim3(32, 8), 0, stream>>>(V, Vt);
    energy_topk_kernel<<<NROWS / 32, 256, 0, stream>>>(h_all, out, idx_int, partial,
                                                       captured_row, counts);
    xhat_wmma_kernel<<<NROWS, 256, 0, stream>>>(x_flat, out, idx_int, Vt,
                                                uncap_row, recon_row);
    colsum_reduce_kernel<<<MEXP / 256, 256, 0, stream>>>(partial, avg);
    finalize_kernel<<<1, 256, 0, stream>>>(captured_row, uncap_row, recon_row,
                                           avg, counts, out);
}